// PAMNetModel_33818572488720
// MI455X (gfx1250) — compile-verified
//
#include <hip/hip_runtime.h>
#include <math.h>

typedef float v2f __attribute__((ext_vector_type(2)));
typedef float v8f __attribute__((ext_vector_type(8)));

#define HID 128

// ---------------------------------------------------------------- utilities

__global__ void zero_f32(float* __restrict__ p, long long n) {
  long long i = (long long)blockIdx.x * blockDim.x + threadIdx.x;
  long long stride = (long long)gridDim.x * blockDim.x;
  for (; i < n; i += stride) p[i] = 0.0f;
}

// ------------------------------------------------- node GEMM: Y = X@W (+b)
// block = 256 threads = 8 waves; block covers 16 rows x 128 cols.
// Each wave computes one 16x16 tile with V_WMMA_F32_16X16X4_F32 (32 steps).
__global__ __launch_bounds__(256) void node_gemm(const float* __restrict__ X,
                                                 const float* __restrict__ W,
                                                 const float* __restrict__ bias,
                                                 float* __restrict__ Y, int N) {
  __shared__ float As[16 * 132];
  const int t = threadIdx.x;
  const int row0 = blockIdx.x * 16;
  // stage 16x128 A tile in LDS (stride 132 -> conflict-free fragment reads)
  for (int i = 0; i < 8; ++i) {
    int idx = t + i * 256;
    int r = idx >> 7, c = idx & 127;
    int rr = row0 + r; if (rr >= N) rr = N - 1;
    As[r * 132 + c] = X[(long long)rr * HID + c];
  }
  __syncthreads();

  const int wave = t >> 5, lane = t & 31;
  const int half = lane >> 4, m = lane & 15;
  const int colbase = wave * 16;
  v8f acc = {};
  for (int k = 0; k < HID; k += 4) {
    v2f a, b;
    // 32-bit A 16x4 layout: lanes0-15 hold K=k,k+1 ; lanes16-31 hold K=k+2,k+3
    a.x = As[m * 132 + k + 2 * half];
    a.y = As[m * 132 + k + 1 + 2 * half];
    b.x = W[(k + 2 * half) * HID + colbase + m];
    b.y = W[(k + 1 + 2 * half) * HID + colbase + m];
    acc = __builtin_amdgcn_wmma_f32_16x16x4_f32(false, a, false, b, (short)0,
                                                acc, false, false);
  }
  const int col = colbase + m;
  const float bv = bias ? bias[col] : 0.0f;
  for (int v = 0; v < 8; ++v) {
    int r = row0 + v + 8 * half;
    if (r < N) Y[(long long)r * HID + col] = acc[v] + bv;
  }
}

// -------------------------------------------------------- degree / dinv / counts

__global__ void degree_kernel(const int* __restrict__ dst, float* __restrict__ deg,
                              int E) {
  int i = blockIdx.x * blockDim.x + threadIdx.x;
  if (i < E) atomicAdd(&deg[dst[i]], 1.0f);
}

__global__ void dinv_kernel(const float* __restrict__ deg, float* __restrict__ dinv,
                            int N) {
  int i = blockIdx.x * blockDim.x + threadIdx.x;
  if (i < N) dinv[i] = rsqrtf(deg[i] + 1.0f);  // +1 = self loop; always > 0
}

__global__ void count_kernel(const int* __restrict__ batch, float* __restrict__ counts,
                             int N) {
  int i = blockIdx.x * blockDim.x + threadIdx.x;
  if (i < N) atomicAdd(&counts[batch[i]], 1.0f);
}

// --------------------------------------- GCN edge scatter: gcn[d] += n*xw[s]
// one thread handles one (edge, 4-channel) group

__global__ void gcn_scatter(const int* __restrict__ src, const int* __restrict__ dst,
                            const float* __restrict__ dinv,
                            const float* __restrict__ xw, float* __restrict__ gcn,
                            int E) {
  long long gid = (long long)blockIdx.x * blockDim.x + threadIdx.x;
  int e = (int)(gid >> 5);
  if (e >= E) return;
  int c4 = ((int)gid & 31) << 2;
  int s = src[e], d = dst[e];
  float nrm = dinv[s] * dinv[d];
  const float4 v = *(const float4*)(xw + (long long)s * HID + c4);
  float* g = gcn + (long long)d * HID + c4;
  atomicAdd(g + 0, nrm * v.x);
  atomicAdd(g + 1, nrm * v.y);
  atomicAdd(g + 2, nrm * v.z);
  atomicAdd(g + 3, nrm * v.w);
}

// --------------------------------------------- PAMConv edge kernel (16 edges/block)
// h = relu(delta@pm_w1+b1) in LDS, de = h@pm_w2+b2 via WMMA,
// att = sigmoid([x_i,x_j]@att_w+b), msg = att*(x_j+de) scattered to loc[dst].

__global__ __launch_bounds__(256) void edge_kernel(
    const int* __restrict__ src, const int* __restrict__ dst,
    const float* __restrict__ pos, const float* __restrict__ xl,
    const float* __restrict__ pm_w1, const float* __restrict__ pm_b1,
    const float* __restrict__ pm_w2, const float* __restrict__ pm_b2,
    const float* __restrict__ att_w, const float* __restrict__ att_b,
    float* __restrict__ loc, int E) {
  __shared__ float Hs[16 * 132];
  __shared__ float att_s[16];
  __shared__ float dlt[16][3];
  __shared__ int ssrc[16], sdst[16];
  const int t = threadIdx.x;
  const int e0 = blockIdx.x * 16;

  if (t < 16) {
    int e = e0 + t; if (e >= E) e = E - 1;
    ssrc[t] = src[e];
    sdst[t] = dst[e];
    att_s[t] = att_b[0];
  }
  __syncthreads();
  if (t < 48) {
    int er = t / 3, j = t % 3;
    dlt[er][j] = pos[(long long)sdst[er] * 3 + j] - pos[(long long)ssrc[er] * 3 + j];
  }
  __syncthreads();

  // first MLP layer: Hs[er][c] = relu(delta . pm_w1[:,c] + b1[c])
  for (int i = 0; i < 8; ++i) {
    int idx = t + i * 256;
    int er = idx >> 7, c = idx & 127;
    float v = dlt[er][0] * pm_w1[c] + dlt[er][1] * pm_w1[HID + c] +
              dlt[er][2] * pm_w1[2 * HID + c] + pm_b1[c];
    Hs[er * 132 + c] = fmaxf(v, 0.0f);
  }
  // attention logits: 16 threads per edge, each sums a stride-16 slice
  {
    int er = t >> 4, l16 = t & 15;
    const float* xi = xl + (long long)sdst[er] * HID;  // x_i = xl[dst]
    const float* xj = xl + (long long)ssrc[er] * HID;  // x_j = xl[src]
    float p = 0.0f;
    for (int kk = l16; kk < HID; kk += 16)
      p += xi[kk] * att_w[kk] + xj[kk] * att_w[HID + kk];
    atomicAdd(&att_s[er], p);  // ds_add_f32
  }
  __syncthreads();
  if (t < 16) att_s[t] = 1.0f / (1.0f + expf(-att_s[t]));
  __syncthreads();

  // second MLP layer via WMMA: de(16x16 tile per wave) = Hs @ pm_w2
  const int wave = t >> 5, lane = t & 31;
  const int half = lane >> 4, m = lane & 15;
  const int colbase = wave * 16;
  v8f acc = {};
  for (int k = 0; k < HID; k += 4) {
    v2f a, b;
    a.x = Hs[m * 132 + k + 2 * half];
    a.y = Hs[m * 132 + k + 1 + 2 * half];
    b.x = pm_w2[(k + 2 * half) * HID + colbase + m];
    b.y = pm_w2[(k + 1 + 2 * half) * HID + colbase + m];
    acc = __builtin_amdgcn_wmma_f32_16x16x4_f32(false, a, false, b, (short)0,
                                                acc, false, false);
  }
  const int col = colbase + m;
  const float b2 = pm_b2[col];
  for (int v = 0; v < 8; ++v) {
    int er = v + 8 * half;
    if (e0 + er < E) {
      float de = acc[v] + b2;
      float msg = att_s[er] * (xl[(long long)ssrc[er] * HID + col] + de);
      atomicAdd(&loc[(long long)sdst[er] * HID + col], msg);
    }
  }
}

// --------------------------- fusion (K=256 WMMA) + pooled accumulation
// A row = relu([gcn + dinv^2*xw + gcn_b  ||  loc]); result accumulated into
// pooled[batch[n]] directly (fused activations never stored).

__global__ __launch_bounds__(256) void fusion_pool_kernel(
    const float* __restrict__ gcn, const float* __restrict__ xw,
    const float* __restrict__ loc, const float* __restrict__ dinv,
    const float* __restrict__ gcn_b, const float* __restrict__ fus_w,
    const float* __restrict__ fus_b, const int* __restrict__ batch,
    float* __restrict__ pooled, int N) {
  __shared__ float As[16 * 260];
  __shared__ int sb[16];
  const int t = threadIdx.x;
  const int n0 = blockIdx.x * 16;
  if (t < 16) {
    int n = n0 + t; if (n >= N) n = N - 1;
    sb[t] = batch[n];
  }
  for (int i = 0; i < 16; ++i) {
    int idx = t + i * 256;
    int r = idx >> 8, k = idx & 255;
    int n = n0 + r; if (n >= N) n = N - 1;
    float v;
    if (k < HID) {
      float di = dinv[n];
      v = gcn[(long long)n * HID + k] + di * di * xw[(long long)n * HID + k] +
          gcn_b[k];
    } else {
      v = loc[(long long)n * HID + (k - HID)];
    }
    As[r * 260 + k] = fmaxf(v, 0.0f);
  }
  __syncthreads();

  const int wave = t >> 5, lane = t & 31;
  const int half = lane >> 4, m = lane & 15;
  const int colbase = wave * 16;
  v8f acc = {};
  for (int k = 0; k < 2 * HID; k += 4) {
    v2f a, b;
    a.x = As[m * 260 + k + 2 * half];
    a.y = As[m * 260 + k + 1 + 2 * half];
    b.x = fus_w[(k + 2 * half) * HID + colbase + m];
    b.y = fus_w[(k + 1 + 2 * half) * HID + colbase + m];
    acc = __builtin_amdgcn_wmma_f32_16x16x4_f32(false, a, false, b, (short)0,
                                                acc, false, false);
  }
  const int col = colbase + m;
  const float fb = fus_b[col];
  for (int v = 0; v < 8; ++v) {
    int r = v + 8 * half;
    if (n0 + r < N) {
      float val = fmaxf(acc[v] + fb, 0.0f);
      atomicAdd(&pooled[(long long)sb[r] * HID + col], val);
    }
  }
}

// ------------------------------------------------ per-graph readout MLP

__global__ __launch_bounds__(64) void readout_kernel(
    const float* __restrict__ pooled, const float* __restrict__ counts,
    const float* __restrict__ l1_w, const float* __restrict__ l1_b,
    const float* __restrict__ l2_w, const float* __restrict__ l2_b,
    float* __restrict__ out) {
  __shared__ float red[64];
  __shared__ float pm[128];
  const int g = blockIdx.x, t = threadIdx.x;
  const float inv = 1.0f / fmaxf(counts[g], 1.0f);
  pm[t] = pooled[g * 128 + t] * inv;
  pm[t + 64] = pooled[g * 128 + t + 64] * inv;
  __syncthreads();
  float h = l1_b[t];
  for (int k = 0; k < 128; ++k) h += pm[k] * l1_w[k * 64 + t];
  h = fmaxf(h, 0.0f);
  red[t] = h * l2_w[t];
  __syncthreads();
  for (int s = 32; s > 0; s >>= 1) {
    if (t < s) red[t] += red[t + s];
    __syncthreads();
  }
  if (t == 0) out[g] = red[0] + l2_b[0];
}

// --------------------------------------------------------------- launcher

extern "C" void kernel_launch(void* const* d_in, const int* in_sizes, int n_in,
                              void* d_out, int out_size, void* d_ws, size_t ws_size,
                              hipStream_t stream) {
  (void)n_in; (void)ws_size;
  const float* x     = (const float*)d_in[0];
  const float* pos   = (const float*)d_in[1];
  const int*   eidx  = (const int*)d_in[2];
  const int*   batch = (const int*)d_in[3];
  const float* gcn_w = (const float*)d_in[4];
  const float* gcn_b = (const float*)d_in[5];
  const float* pam_w = (const float*)d_in[6];
  const float* pam_b = (const float*)d_in[7];
  const float* pm_w1 = (const float*)d_in[8];
  const float* pm_b1 = (const float*)d_in[9];
  const float* pm_w2 = (const float*)d_in[10];
  const float* pm_b2 = (const float*)d_in[11];
  const float* att_w = (const float*)d_in[12];
  const float* att_b = (const float*)d_in[13];
  const float* fus_w = (const float*)d_in[14];
  const float* fus_b = (const float*)d_in[15];
  const float* l1_w  = (const float*)d_in[16];
  const float* l1_b  = (const float*)d_in[17];
  const float* l2_w  = (const float*)d_in[18];
  const float* l2_b  = (const float*)d_in[19];
  float* out = (float*)d_out;

  const int N = in_sizes[0] / HID;
  const int E = in_sizes[2] / 2;
  const int G = out_size;

  const int* src = eidx;
  const int* dst = eidx + E;

  float* ws = (float*)d_ws;
  long long off = 0;
  float* xw     = ws + off; off += (long long)N * HID;
  float* xl     = ws + off; off += (long long)N * HID;
  float* gcn    = ws + off; off += (long long)N * HID;  // zeroed
  float* loc    = ws + off; off += (long long)N * HID;  // zeroed
  float* deg    = ws + off; off += N;                   // zeroed
  float* pooled = ws + off; off += (long long)G * HID;  // zeroed
  float* counts = ws + off; off += G;                   // zeroed
  float* dinv   = ws + off; off += N;

  // zero the contiguous accumulator span: gcn..counts
  long long zlen = 2LL * N * HID + N + (long long)G * HID + G;
  zero_f32<<<1024, 256, 0, stream>>>(gcn, zlen);

  const int nb16 = (N + 15) / 16;
  node_gemm<<<nb16, 256, 0, stream>>>(x, gcn_w, nullptr, xw, N);
  node_gemm<<<nb16, 256, 0, stream>>>(x, pam_w, pam_b, xl, N);

  degree_kernel<<<(E + 255) / 256, 256, 0, stream>>>(dst, deg, E);
  dinv_kernel<<<(N + 255) / 256, 256, 0, stream>>>(deg, dinv, N);
  count_kernel<<<(N + 255) / 256, 256, 0, stream>>>(batch, counts, N);

  long long sthreads = (long long)E * 32;
  gcn_scatter<<<(unsigned)((sthreads + 255) / 256), 256, 0, stream>>>(
      src, dst, dinv, xw, gcn, E);

  edge_kernel<<<(E + 15) / 16, 256, 0, stream>>>(src, dst, pos, xl, pm_w1, pm_b1,
                                                 pm_w2, pm_b2, att_w, att_b, loc, E);

  fusion_pool_kernel<<<nb16, 256, 0, stream>>>(gcn, xw, loc, dinv, gcn_b, fus_w,
                                               fus_b, batch, pooled, N);

  readout_kernel<<<G, 64, 0, stream>>>(pooled, counts, l1_w, l1_b, l2_w, l2_b, out);
}